// FastSurfaceDiceLoss2D_34308198760619
// MI455X (gfx1250) — compile-verified
//
#include <hip/hip_runtime.h>
#include <stdint.h>

#define W     512
#define OW    511
#define BAND  32
#define NB    16      // ceil(511/32)
#define NIMG  64
#define PADW  520     // padded LDS row (even -> float2-aligned, room for col 512 read)

typedef __attribute__((ext_vector_type(2))) float v2f;
typedef __attribute__((ext_vector_type(8))) float v8f;

// lp = log_sigmoid(x), ln = log_sigmoid(-x) = lp - x   (one exp + one log total)
__device__ __forceinline__ void logsig2(float x, float& lp, float& ln) {
    float e = __expf(-fabsf(x));
    float l = __logf(1.0f + e);
    lp = fminf(x, 0.0f) - l;
    ln = lp - x;
}

__device__ __forceinline__ void accum(float lps, float lns,
                                      float tl, float tr, float bl, float br,
                                      float s[8]) {
    float fg   = __expf(lps);            // lps <= 0
    float bg   = __expf(lns);            // lns <= 0
    float surf = fmaxf(0.0f, 1.0f - fg - bg);
    unsigned code = (tl > 0.5f ? 1u : 0u) | (tr > 0.5f ? 2u : 0u) |
                    (bl > 0.5f ? 4u : 0u) | (br > 0.5f ? 8u : 0u);
    const unsigned MZ = 0x8001u;  // codes {0,15}      -> 0
    const unsigned MD = 0x0240u;  // codes {6,9}       -> sqrt(2)
    const unsigned MS = 0x1428u;  // codes {3,5,10,12} -> 1
    float contour;
    if      ((MZ >> code) & 1u) contour = 0.0f;
    else if ((MD >> code) & 1u) contour = 1.41421356f;
    else if ((MS >> code) & 1u) contour = 1.0f;
    else                        contour = 0.70710678f;
    float gts  = (contour > 0.0f) ? 1.0f : 0.0f;
    float isfg = (code == 15u) ? 1.0f : 0.0f;
    float isbg = (code == 0u)  ? 1.0f : 0.0f;
    s[0] += fg * isfg;
    s[1] += fg;
    s[2] += bg * isbg;
    s[3] += bg;
    s[4] += surf * contour;
    s[5] += (surf + gts) * contour;
    s[6] += isfg;
    s[7] += isbg;
}

__global__ void __launch_bounds__(256) surf_dice_partial(
        const float* __restrict__ preds, const float* __restrict__ targets,
        float* __restrict__ ws) {
    __shared__ float s_lp[2][PADW];
    __shared__ float s_ln[2][PADW];
    __shared__ float s_tt[2][PADW];
    __shared__ float s_red[64];

    const int t    = threadIdx.x;
    const int img  = blockIdx.y;
    const int band = blockIdx.x;
    const int r0   = band * BAND;
    const int rows = min(BAND, OW - r0);

    const float* pbase = preds   + (size_t)img * (W * W);
    const float* tbase = targets + (size_t)img * (W * W);

    const int c0 = 2 * t;
    // LDS byte addresses for async-to-LDS (low 32 bits of generic LDS pointer)
    const uint32_t tt_lds0 = (uint32_t)(uintptr_t)&s_tt[0][c0];
    const uint32_t tt_lds1 = (uint32_t)(uintptr_t)&s_tt[1][c0];

    // ---- preload input row r0 ----
    {
        uint32_t goff = (uint32_t)((r0 * W + c0) * 4);
        asm volatile("global_load_async_to_lds_b64 %0, %1, %2"
                     :: "v"(tt_lds0), "v"(goff), "s"(tbase) : "memory");
        float2 p = *(const float2*)(pbase + (size_t)r0 * W + c0);
        float lp0, ln0, lp1, ln1;
        logsig2(p.x, lp0, ln0);
        logsig2(p.y, lp1, ln1);
        s_lp[0][c0] = lp0; s_lp[0][c0 + 1] = lp1;
        s_ln[0][c0] = ln0; s_ln[0][c0 + 1] = ln1;
    }
    asm volatile("s_wait_asynccnt 0" ::: "memory");
    __syncthreads();

    float s[8] = {0.f, 0.f, 0.f, 0.f, 0.f, 0.f, 0.f, 0.f};
    int cur = 0;
    for (int k = 0; k < rows; ++k) {
        const int nxt = cur ^ 1;
        const int rin = r0 + k + 1;

        // async-stream next targets row straight into LDS (double-buffered)
        uint32_t goff = (uint32_t)((rin * W + c0) * 4);
        uint32_t lds  = nxt ? tt_lds1 : tt_lds0;
        asm volatile("global_load_async_to_lds_b64 %0, %1, %2"
                     :: "v"(lds), "v"(goff), "s"(tbase) : "memory");

        // next preds row: load, transform once, share via LDS
        float2 p = *(const float2*)(pbase + (size_t)rin * W + c0);
        float lpa, lna, lpb, lnb;
        logsig2(p.x, lpa, lna);
        logsig2(p.y, lpb, lnb);
        s_lp[nxt][c0] = lpa; s_lp[nxt][c0 + 1] = lpb;
        s_ln[nxt][c0] = lna; s_ln[nxt][c0 + 1] = lnb;

        asm volatile("s_wait_asynccnt 0" ::: "memory");
        __syncthreads();

        float2 lpA = *(const float2*)&s_lp[cur][c0];  float lpA2 = s_lp[cur][c0 + 2];
        float2 lpB = *(const float2*)&s_lp[nxt][c0];  float lpB2 = s_lp[nxt][c0 + 2];
        float2 lnA = *(const float2*)&s_ln[cur][c0];  float lnA2 = s_ln[cur][c0 + 2];
        float2 lnB = *(const float2*)&s_ln[nxt][c0];  float lnB2 = s_ln[nxt][c0 + 2];
        float2 ttA = *(const float2*)&s_tt[cur][c0];  float ttA2 = s_tt[cur][c0 + 2];
        float2 ttB = *(const float2*)&s_tt[nxt][c0];  float ttB2 = s_tt[nxt][c0 + 2];

        accum(lpA.x + lpA.y + lpB.x + lpB.y,
              lnA.x + lnA.y + lnB.x + lnB.y,
              ttA.x, ttA.y, ttB.x, ttB.y, s);
        if (c0 + 1 < OW) {
            accum(lpA.y + lpA2 + lpB.y + lpB2,
                  lnA.y + lnA2 + lnB.y + lnB2,
                  ttA.y, ttA2, ttB.y, ttB2, s);
        }
        __syncthreads();   // protect cur buffers before next iteration overwrites
        cur = nxt;
    }

    // ---- deterministic block reduction: wave32 shuffle tree + LDS ----
#pragma unroll
    for (int kk = 0; kk < 8; ++kk) {
#pragma unroll
        for (int m = 16; m >= 1; m >>= 1)
            s[kk] += __shfl_xor(s[kk], m, 32);
    }
    const int wave = t >> 5;
    const int lane = t & 31;
    if (lane == 0) {
#pragma unroll
        for (int kk = 0; kk < 8; ++kk) s_red[wave * 8 + kk] = s[kk];
    }
    __syncthreads();
    if (t < 8) {
        float v = 0.0f;
#pragma unroll
        for (int w = 0; w < 8; ++w) v += s_red[w * 8 + t];
        ws[((size_t)img * NB + band) * 8 + t] = v;
    }
}

__global__ void __launch_bounds__(64) surf_dice_final(
        const float* __restrict__ ws, float* __restrict__ out) {
    __shared__ float dbuf[64];
    const int i = threadIdx.x;           // one image per thread
    float a[8] = {0.f, 0.f, 0.f, 0.f, 0.f, 0.f, 0.f, 0.f};
    const float* p = ws + (size_t)i * NB * 8;
#pragma unroll
    for (int b = 0; b < NB; ++b)
#pragma unroll
        for (int kk = 0; kk < 8; ++kk) a[kk] += p[b * 8 + kk];

    const float EPSF = 1e-5f;
    float fg_dice = (2.0f * a[0] + EPSF) / (a[1] + a[6] + EPSF);
    float bg_dice = (2.0f * a[2] + EPSF) / (a[3] + a[7] + EPSF);
    float sf_dice = (2.0f * a[4] + EPSF) / (a[5] + EPSF);
    dbuf[i] = (fg_dice + bg_dice + sf_dice) * (1.0f / 3.0f);
    __syncthreads();

    // Sum of 64 dice values via V_WMMA_F32_16X16X4_F32:
    // B = all-ones so D[m][n] = sum_k A[m][k]; column-0 sum == total, and the
    // result is layout-invariant for A (every packed element counted once).
    const int lane = i & 31;
    v2f A; A.x = dbuf[2 * lane]; A.y = dbuf[2 * lane + 1];
    v2f B; B.x = 1.0f; B.y = 1.0f;
    v8f C = {};
    v8f D = __builtin_amdgcn_wmma_f32_16x16x4_f32(
        false, A, false, B, (short)0, C, false, false);
    float tsum = D[0] + D[1] + D[2] + D[3] + D[4] + D[5] + D[6] + D[7];
    float total = __shfl(tsum, 0, 32) + __shfl(tsum, 16, 32);
    if (i == 0) out[0] = 1.0f - total * (1.0f / 64.0f);
}

extern "C" void kernel_launch(void* const* d_in, const int* in_sizes, int n_in,
                              void* d_out, int out_size, void* d_ws, size_t ws_size,
                              hipStream_t stream) {
    (void)in_sizes; (void)n_in; (void)out_size; (void)ws_size;
    const float* preds   = (const float*)d_in[0];
    const float* targets = (const float*)d_in[1];
    float* ws = (float*)d_ws;   // NIMG * NB * 8 floats = 32 KB
    dim3 grid(NB, NIMG, 1);
    surf_dice_partial<<<grid, 256, 0, stream>>>(preds, targets, ws);
    surf_dice_final<<<1, 64, 0, stream>>>(ws, (float*)d_out);
}